// CircularMaxPool2d_8194797601069
// MI455X (gfx1250) — compile-verified
//
#include <hip/hip_runtime.h>
#include <cstdint>
#include <cstddef>

// ---------------------------------------------------------------------------
// CircularMaxPool2d, radius=5 (disk mask), reflect padding, NaN->-99 sentinel.
// Input  : [8,1,2048,2048] f32 ; Output: same shape/dtype.
//
// Separable disk decomposition (row half-widths by |dy|: 5,4,4,4,3,0):
//   H3(x) = max raw[x-3..x+3]          (7-tap)
//   H4(x) = max(H3(x-1), H3(x+1))      -> 9-wide window
//   H5(x) = max(H4(x-1), H4(x+1))      -> 11-wide window
//   out(x,y) = max( H5(x,y), H4(x,y±1..±3), H3(x,y±4), raw(x,y±5) )
//
// Data movement: CDNA5 async DMA global->LDS (ASYNCcnt), B128 fast path on
// x-interior tiles, per-element B32 + reflect on x-border tiles.
// ---------------------------------------------------------------------------

#define IMG_H 2048
#define IMG_W 2048
#define NB    8
#define RAD   5

#define TX 64
#define TY 64
#define BLOCK 256

#define RROWS (TY + 2 * RAD)         // 74 rows of raw / H3 / H4
#define XOFF  8                      // raw col lx corresponds to global x = gx0 - XOFF + lx
#define RSTRIDE 80                   // raw row stride (floats); 20 x B128 per row
#define NCOLS 74                     // consumed raw cols: lx in [3, 76]
#define H3C (TX + 4)                 // H3 cols: j in [-2, TX+1], stored at j+2
#define H4C (TX + 2)                 // H4 cols: j in [-1, TX],   stored at j+1
#define HSTRIDE 68

#define RAW_OFF 0
#define H3_OFF  (RROWS * RSTRIDE)                 // 5920
#define H4_OFF  (H3_OFF + RROWS * HSTRIDE)        // 10952
#define SMEM_FLOATS (H4_OFF + RROWS * HSTRIDE)    // 15984 floats = 63936 B

#define NAN_SENT (-99.0f)

// Low 32 bits of a generic LDS pointer are the LDS byte offset
// (hardware truncates addr[31:0] inside the LDS aperture).
__device__ __forceinline__ uint32_t lds_byte_off(const float* p) {
    return (uint32_t)(uintptr_t)p;
}

__device__ __forceinline__ int reflect_idx(int c, int n) {
    // jnp.pad 'reflect' (edge not repeated); one reflection suffices (RAD << n)
    c = (c < 0) ? -c : c;
    c = (c >= n) ? (2 * n - 2 - c) : c;
    return c;
}

__global__ void __launch_bounds__(BLOCK)
circular_maxpool_r5_kernel(const float* __restrict__ x, float* __restrict__ out) {
    extern __shared__ float smem[];

    const int tid = threadIdx.x;
    const int gx0 = blockIdx.x * TX;
    const int gy0 = blockIdx.y * TY;
    const int b   = blockIdx.z;

    const float* img = x + (size_t)b * IMG_H * IMG_W;

    // ---- Phase 1: async DMA halo tile (global -> LDS) ------------------------
    // Rows are contiguous in memory even when y-reflected, so x-interior tiles
    // can use 16B-aligned B128 DMA (read window [gx0-8, gx0+72), gx0 % 64 == 0).
    const bool xInterior = (gx0 >= XOFF) && (gx0 + RSTRIDE - XOFF <= IMG_W);

    if (xInterior) {
        for (int e = tid; e < RROWS * 20; e += BLOCK) {
            const int ly = e / 20;
            const int q  = e - ly * 20;                    // which B128 in the row
            const int gy = reflect_idx(gy0 + ly - RAD, IMG_H);
            const uint32_t goff  = (uint32_t)((gy * IMG_W + (gx0 - XOFF) + 4 * q) * 4);
            const uint32_t laddr = lds_byte_off(&smem[RAW_OFF + ly * RSTRIDE + 4 * q]);
            asm volatile("global_load_async_to_lds_b128 %0, %1, %2"
                         :
                         : "v"(laddr), "v"(goff), "s"(img)
                         : "memory");
        }
    } else {
        for (int e = tid; e < RROWS * NCOLS; e += BLOCK) {
            const int ly = e / NCOLS;
            const int c  = e - ly * NCOLS;                 // cols gx0-5 .. gx0+68
            const int gy = reflect_idx(gy0 + ly - RAD, IMG_H);
            const int gx = reflect_idx(gx0 + c - RAD, IMG_W);
            const uint32_t goff  = (uint32_t)((gy * IMG_W + gx) * 4);
            const uint32_t laddr = lds_byte_off(&smem[RAW_OFF + ly * RSTRIDE + c + 3]);
            asm volatile("global_load_async_to_lds_b32 %0, %1, %2"
                         :
                         : "v"(laddr), "v"(goff), "s"(img)
                         : "memory");
        }
    }
    asm volatile("s_wait_asynccnt 0" ::: "memory");
    __syncthreads();

    // ---- Phase 2: NaN -> sentinel, in place, only on consumed cols ----------
    for (int e = tid; e < RROWS * NCOLS; e += BLOCK) {
        const int ly = e / NCOLS;
        const int c  = e - ly * NCOLS;
        const int idx = RAW_OFF + ly * RSTRIDE + c + 3;
        const float v = smem[idx];
        if (v != v) smem[idx] = NAN_SENT;
    }
    __syncthreads();

    // ---- Phase 3: H3 = 7-tap horizontal max ---------------------------------
    // H3 at [row][j2], j2 = j+2 in [0,68): window lx = j2+3 .. j2+9
    for (int e = tid; e < RROWS * H3C; e += BLOCK) {
        const int row = e / H3C;
        const int j2  = e - row * H3C;
        const float* rp = &smem[RAW_OFF + row * RSTRIDE + j2 + 3];
        float m = rp[0];
        m = fmaxf(m, rp[1]);
        m = fmaxf(m, rp[2]);
        m = fmaxf(m, rp[3]);
        m = fmaxf(m, rp[4]);
        m = fmaxf(m, rp[5]);
        m = fmaxf(m, rp[6]);
        smem[H3_OFF + row * HSTRIDE + j2] = m;
    }
    __syncthreads();

    // ---- Phase 4: H4(j) = max(H3(j-1), H3(j+1)), stored at [row][j+1] -------
    for (int e = tid; e < RROWS * H4C; e += BLOCK) {
        const int row = e / H4C;
        const int i   = e - row * H4C;
        const int hb  = H3_OFF + row * HSTRIDE;
        smem[H4_OFF + row * HSTRIDE + i] = fmaxf(smem[hb + i], smem[hb + i + 2]);
    }
    __syncthreads();

    // ---- Phase 5: vertical combine + sentinel -> NaN + NT store -------------
    for (int e = tid; e < TX * TY; e += BLOCK) {
        const int oy = e / TX;
        const int ox = e - oy * TX;
        const int ry = oy + RAD;  // row index into raw/H3/H4 space

        // H5 at center row: max(H4(x-1), H4(x+1)) -> h4[ry][ox], h4[ry][ox+2]
        float m = fmaxf(smem[H4_OFF + ry * HSTRIDE + ox],
                        smem[H4_OFF + ry * HSTRIDE + ox + 2]);
        // |dy| = 1..3 : H4(x) -> h4[.][ox+1]
        m = fmaxf(m, smem[H4_OFF + (ry - 1) * HSTRIDE + ox + 1]);
        m = fmaxf(m, smem[H4_OFF + (ry + 1) * HSTRIDE + ox + 1]);
        m = fmaxf(m, smem[H4_OFF + (ry - 2) * HSTRIDE + ox + 1]);
        m = fmaxf(m, smem[H4_OFF + (ry + 2) * HSTRIDE + ox + 1]);
        m = fmaxf(m, smem[H4_OFF + (ry - 3) * HSTRIDE + ox + 1]);
        m = fmaxf(m, smem[H4_OFF + (ry + 3) * HSTRIDE + ox + 1]);
        // |dy| = 4 : H3(x) -> h3[.][ox+2]
        m = fmaxf(m, smem[H3_OFF + (ry - 4) * HSTRIDE + ox + 2]);
        m = fmaxf(m, smem[H3_OFF + (ry + 4) * HSTRIDE + ox + 2]);
        // |dy| = 5 : raw(x) -> raw[.][ox+XOFF]
        m = fmaxf(m, smem[RAW_OFF + (ry - 5) * RSTRIDE + ox + XOFF]);
        m = fmaxf(m, smem[RAW_OFF + (ry + 5) * RSTRIDE + ox + XOFF]);

        const float o = (m == NAN_SENT) ? __builtin_nanf("") : m;
        // Streaming write-once output: non-temporal so it doesn't evict
        // input halo lines from the 192MB L2.
        __builtin_nontemporal_store(o,
            &out[((size_t)b * IMG_H + (gy0 + oy)) * IMG_W + (gx0 + ox)]);
    }
}

extern "C" void kernel_launch(void* const* d_in, const int* in_sizes, int n_in,
                              void* d_out, int out_size, void* d_ws, size_t ws_size,
                              hipStream_t stream) {
    (void)in_sizes; (void)n_in; (void)out_size; (void)d_ws; (void)ws_size;
    const float* x  = (const float*)d_in[0];   // [8,1,2048,2048] f32
    // d_in[1] is radius (=5), baked in at compile time.
    float* out = (float*)d_out;

    dim3 grid(IMG_W / TX, IMG_H / TY, NB);     // 32 x 32 x 8 blocks
    dim3 block(BLOCK);
    const size_t shmem = SMEM_FLOATS * sizeof(float);  // 63,936 bytes
    circular_maxpool_r5_kernel<<<grid, block, shmem, stream>>>(x, out);
}